// GroupAttention_65146063946605
// MI455X (gfx1250) — compile-verified
//
#include <hip/hip_runtime.h>

// Problem constants (match reference)
#define DM   512            // d_model
#define NSS  128            // spatial sites
#define GSS  8              // group size
#define NH   8              // heads
#define DKK  64             // head dim
#define NB   4              // batch
#define TOK  4096           // NB * NSS * GSS
#define SEQ  1024           // NSS * GSS (per batch)

// LDS layout for the attention kernel (floats). Dynamic LDS base assumed 0
// (no static __shared__ in that kernel).
#define PST  1028                     // logits row stride (bank-spread)
#define VS_STR 68                     // V-chunk row stride (16B-aligned rows)
#define QS_STR 66                     // Q-tile row stride (TDM pad 2dw/64dw)
#define VS_OFF (16 * PST)             // 16448
#define QS_OFF (VS_OFF + 128 * VS_STR) // 25152
#define SMEM_FLOATS (QS_OFF + 16 * QS_STR) // 26208 -> 104832 bytes

typedef float v2f __attribute__((ext_vector_type(2)));
typedef float v8f __attribute__((ext_vector_type(8)));
typedef unsigned int u32x4 __attribute__((ext_vector_type(4)));
typedef int i32x4 __attribute__((ext_vector_type(4)));
typedef int i32x8 __attribute__((ext_vector_type(8)));

// CDNA5 fp32 matrix core: D(16x16,f32) = A(16x4,f32) * B(4x16,f32) + C
__device__ __forceinline__ v8f wmma4(v2f a, v2f b, v8f c) {
  return __builtin_amdgcn_wmma_f32_16x16x4_f32(false, a, false, b, (short)0, c,
                                               false, false);
}

// ---------------------------------------------------------------------------
// Kernel 1: LayerNorm over last dim (512). One block (256 thr) per token.
// ---------------------------------------------------------------------------
__global__ __launch_bounds__(256) void ln_kernel(const float* __restrict__ X,
                                                 const float* __restrict__ g,
                                                 const float* __restrict__ be,
                                                 float* __restrict__ Xn) {
  __shared__ float redA[8];
  __shared__ float redB[8];
  const int t   = blockIdx.x;
  const int tid = threadIdx.x;
  const float* x = X + (size_t)t * DM;

  float v0 = x[tid], v1 = x[tid + 256];
  float s = v0 + v1;
#pragma unroll
  for (int off = 16; off; off >>= 1) s += __shfl_xor(s, off, 32);
  if ((tid & 31) == 0) redA[tid >> 5] = s;
  __syncthreads();
  float tot = 0.f;
#pragma unroll
  for (int i = 0; i < 8; ++i) tot += redA[i];
  const float mu = tot * (1.0f / (float)DM);

  const float d0 = v0 - mu, d1 = v1 - mu;
  float q = d0 * d0 + d1 * d1;
#pragma unroll
  for (int off = 16; off; off >>= 1) q += __shfl_xor(q, off, 32);
  if ((tid & 31) == 0) redB[tid >> 5] = q;
  __syncthreads();
  float vt = 0.f;
#pragma unroll
  for (int i = 0; i < 8; ++i) vt += redB[i];
  const float inv = rsqrtf(vt * (1.0f / (float)DM) + 1e-5f);

  Xn[(size_t)t * DM + tid]       = d0 * inv * g[tid]       + be[tid];
  Xn[(size_t)t * DM + tid + 256] = d1 * inv * g[tid + 256] + be[tid + 256];
}

// ---------------------------------------------------------------------------
// Kernel 2: QKV projection. C[t,o] = Xn[t,:] . W[o,:] + bias[o]
// grid = (TOK/128, DM/64, 3); block = 256 (8 waves); wave owns 16x64 tile.
// ---------------------------------------------------------------------------
__global__ __launch_bounds__(256) void qkv_gemm(const float* __restrict__ Xn,
                                                const float* __restrict__ Wq,
                                                const float* __restrict__ bq,
                                                const float* __restrict__ Wk,
                                                const float* __restrict__ bk,
                                                const float* __restrict__ Wv,
                                                const float* __restrict__ bv,
                                                float* __restrict__ Q,
                                                float* __restrict__ K,
                                                float* __restrict__ V) {
  const int lane = threadIdx.x & 31;
  const int wave = threadIdx.x >> 5;
  const int hl   = lane >> 4;    // K-pair half
  const int mr   = lane & 15;    // M (A) / N (B,C) within tile

  const float* W; const float* bias; float* out;
  if (blockIdx.z == 0)      { W = Wq; bias = bq; out = Q; }
  else if (blockIdx.z == 1) { W = Wk; bias = bk; out = K; }
  else                      { W = Wv; bias = bv; out = V; }

  const int mbase = (blockIdx.x * 8 + wave) * 16;
  const int nbase = blockIdx.y * 64;

  v8f acc[4] = {};
  const float* arow = Xn + (size_t)(mbase + mr) * DM + 2 * hl;
  const float* brow = W  + (size_t)(nbase + mr) * DM + 2 * hl;

  // CDNA5 global_prefetch_b8: warm WGP$/L2 for the streamed rows.
  __builtin_prefetch(arow, 0, 3);
#pragma unroll
  for (int nt = 0; nt < 4; ++nt) __builtin_prefetch(brow + nt * 16 * DM, 0, 3);

  for (int k = 0; k < DM; k += 4) {
    v2f a = *(const v2f*)(arow + k);
#pragma unroll
    for (int nt = 0; nt < 4; ++nt) {
      v2f b = *(const v2f*)(brow + (size_t)nt * 16 * DM + k);
      acc[nt] = wmma4(a, b, acc[nt]);
    }
  }

#pragma unroll
  for (int nt = 0; nt < 4; ++nt) {
    const int col = nbase + nt * 16 + mr;
    const float bb = bias[col];
#pragma unroll
    for (int r = 0; r < 8; ++r) {
      const int row = mbase + r + 8 * hl;
      out[(size_t)row * DM + col] = acc[nt][r] + bb;
    }
  }
}

// ---------------------------------------------------------------------------
// Kernel 3: attention per (batch, head, 16-query tile).
//  - Q tile staged into LDS via Tensor Data Mover (async tensor DMA)
//  - logits (16x1024) via WMMA -> bias/scale -> LDS -> softmax in LDS
//  - PV via WMMA with V staged in LDS in 128-row chunks (coalesced b128)
// grid = (SEQ/16, NH, NB); block = 256; dynamic LDS = SMEM_FLOATS*4.
// ---------------------------------------------------------------------------
extern __shared__ float smem[];

__global__ __launch_bounds__(256) void attn_kernel(const float* __restrict__ Q,
                                                   const float* __restrict__ K,
                                                   const float* __restrict__ V,
                                                   const int* __restrict__ sdiff,
                                                   const int* __restrict__ gmult,
                                                   const float* __restrict__ rho,
                                                   float* __restrict__ O) {
  const int qt   = blockIdx.x;            // query tile (16 rows)
  const int m    = blockIdx.y;            // head
  const int b    = blockIdx.z;            // batch
  const int lane = threadIdx.x & 31;
  const int wave = threadIdx.x >> 5;
  const int hl   = lane >> 4;
  const int mr   = lane & 15;
  const size_t rowbase = (size_t)b * SEQ; // token row base for this batch
  const int mcol = m * DKK;

  float* P  = smem;                       // [16][PST]    logits/probs
  float* Vs = smem + VS_OFF;              // [128][VS_STR] V chunk
  float* Qs = smem + QS_OFF;              // [16][QS_STR]  Q tile

  // ---- stage Q tile (16 x 64 f32) into LDS ----
#if __has_builtin(__builtin_amdgcn_tensor_load_to_lds)
  if (wave == 0) {
    const float* qsrc = Q + (rowbase + (size_t)qt * 16) * DM + mcol;
    const unsigned long long ga = (unsigned long long)(uintptr_t)qsrc;
    // D# group 0 (ISA 8.3): count=1 | lds_addr | global_addr | type=2
    u32x4 g0;
    g0.x = 1u;
    g0.y = (unsigned)(QS_OFF * 4);                    // LDS byte address
    g0.z = (unsigned)(ga & 0xFFFFFFFFu);              // global_addr[31:0]
    g0.w = (unsigned)((ga >> 32) & 0x01FFFFFFull) | (2u << 30); // [56:32]|type
    // D# group 1 (ISA 8.4): data_size=4B, pad 2 dwords every 64 dwords
    // (-> LDS row stride 66 floats), tensor 64x16, tile 64x16, stride0=512.
    i32x8 g1;
    g1[0] = (2 << 16) | (1 << 20) | (5 << 22) | (1 << 25);
    g1[1] = (int)(64u << 16);    // tensor_dim0[15:0]  (bits 79:48 lo half)
    g1[2] = (int)(16u << 16);    // tensor_dim1[15:0]  (bits 111:80 lo half)
    g1[3] = (int)(64u << 16);    // tile_dim0 = 64     (bits 127:112)
    g1[4] = 16;                  // tile_dim1 = 16, tile_dim2 = 0
    g1[5] = DM;                  // tensor_dim0_stride[31:0] = 512
    g1[6] = 0;                   // stride0 hi / stride1 lo
    g1[7] = 0;
    i32x4 gz4 = {0, 0, 0, 0};                  // groups 2/3 unused (2D)
    i32x8 gz8 = {0, 0, 0, 0, 0, 0, 0, 0};      // trailing group (clang-23 ABI)
    __builtin_amdgcn_tensor_load_to_lds(g0, g1, gz4, gz4, gz8, 0);
    __builtin_amdgcn_s_wait_tensorcnt(0);             // TENSORcnt == 0
  }
  __syncthreads();
#else
  {
    const int r = threadIdx.x >> 4;           // 0..15
    const int c = (threadIdx.x & 15) * 4;     // 0..60
    const float* src = Q + (rowbase + qt * 16 + r) * DM + mcol + c;
    float* dst = Qs + r * QS_STR + c;
    *(v2f*)(dst)     = *(const v2f*)(src);
    *(v2f*)(dst + 2) = *(const v2f*)(src + 2);
  }
  __syncthreads();
#endif

  // ---- logits: wave handles 8 key-column tiles of 16 ----
  v8f acc[8] = {};
  const float* qr = Qs + mr * QS_STR + 2 * hl;
  for (int k = 0; k < DKK; k += 4) {
    v2f a = *(const v2f*)(qr + k);
#pragma unroll
    for (int ct = 0; ct < 8; ++ct) {
      const int colbase = (wave * 8 + ct) * 16;
      v2f bb = *(const v2f*)(K + (rowbase + colbase + mr) * DM + mcol + k + 2 * hl);
      acc[ct] = wmma4(a, bb, acc[ct]);
    }
  }

  // ---- epilogue: scale + relative-group bias, write to LDS ----
  const float scale = 0.125f;               // 1/sqrt(64)
#pragma unroll
  for (int ct = 0; ct < 8; ++ct) {
    const int tk = (wave * 8 + ct) * 16 + mr;
    const int j  = tk >> 3, kk = tk & 7;
#pragma unroll
    for (int r = 0; r < 8; ++r) {
      const int lr = r + 8 * hl;
      const int tq = qt * 16 + lr;
      const int i = tq >> 3, g = tq & 7;
      const int idx2 = sdiff[i * NSS + j] * GSS + gmult[g * GSS + kk];
      P[lr * PST + tk] = acc[ct][r] * scale + rho[idx2 * NH + m];
    }
  }
  __syncthreads();

  // ---- softmax over 1024 keys, 2 rows per wave ----
#pragma unroll
  for (int rr = 0; rr < 2; ++rr) {
    const int lr = wave * 2 + rr;
    float* row = P + lr * PST;
    float mx = -3.4e38f;
    for (int c = lane; c < SEQ; c += 32) mx = fmaxf(mx, row[c]);
#pragma unroll
    for (int off = 16; off; off >>= 1) mx = fmaxf(mx, __shfl_xor(mx, off, 32));
    float sum = 0.f;
    for (int c = lane; c < SEQ; c += 32) {
      float e = __expf(row[c] - mx);
      row[c] = e;
      sum += e;
    }
#pragma unroll
    for (int off = 16; off; off >>= 1) sum += __shfl_xor(sum, off, 32);
    const float invs = 1.0f / sum;
    for (int c = lane; c < SEQ; c += 32) row[c] *= invs;
  }

  // ---- PV: V staged in LDS chunks of 128 rows; wave -> (ntile, K-sub) ----
  const int nt = wave & 3;                  // 16-col tile of the 64-wide O
  const int ks = wave >> 2;                 // 0/1: 64-row half of each chunk
  v8f o = {};
  for (int ch = 0; ch < 8; ++ch) {
    __syncthreads();                        // P ready / prev chunk consumed
    {
      const int r  = threadIdx.x >> 1;      // 0..127
      const int hf = threadIdx.x & 1;       // 32-float half of a row
      const float* src = V + (rowbase + ch * 128 + r) * DM + mcol + hf * 32;
      float* dst = Vs + r * VS_STR + hf * 32;
#pragma unroll
      for (int u = 0; u < 8; ++u)
        *(float4*)(dst + u * 4) = *(const float4*)(src + u * 4);
    }
    __syncthreads();
    const int kbase = ch * 128 + ks * 64;   // global key index base
    for (int k = 0; k < 64; k += 4) {
      const int kb = kbase + k + 2 * hl;    // global key pair base
      const int rl = ks * 64 + k + 2 * hl;  // row within chunk
      v2f a = *(const v2f*)(P + mr * PST + kb);
      v2f bb;
      bb.x = Vs[rl * VS_STR + nt * 16 + mr];
      bb.y = Vs[(rl + 1) * VS_STR + nt * 16 + mr];
      o = wmma4(a, bb, o);
    }
  }

  // ---- combine the two K-halves and store O ----
  __syncthreads();
  float* red = smem;                        // reuse P region: [8][16][16]
#pragma unroll
  for (int r = 0; r < 8; ++r) red[wave * 256 + (r + 8 * hl) * 16 + mr] = o[r];
  __syncthreads();
  if (ks == 0) {
    const int col = mcol + nt * 16 + mr;
#pragma unroll
    for (int r = 0; r < 8; ++r) {
      const int lr = r + 8 * hl;
      const float val = red[wave * 256 + lr * 16 + mr] +
                        red[(wave + 4) * 256 + lr * 16 + mr];
      O[(rowbase + qt * 16 + lr) * DM + col] = val;
    }
  }
}

// ---------------------------------------------------------------------------
// Kernel 4: output projection + bias + residual. Y = O @ Wo^T + bo + X
// ---------------------------------------------------------------------------
__global__ __launch_bounds__(256) void out_gemm(const float* __restrict__ O,
                                                const float* __restrict__ Wo,
                                                const float* __restrict__ bo,
                                                const float* __restrict__ X,
                                                float* __restrict__ Y) {
  const int lane = threadIdx.x & 31;
  const int wave = threadIdx.x >> 5;
  const int hl   = lane >> 4;
  const int mr   = lane & 15;
  const int mbase = (blockIdx.x * 8 + wave) * 16;
  const int nbase = blockIdx.y * 64;

  v8f acc[4] = {};
  const float* arow = O  + (size_t)(mbase + mr) * DM + 2 * hl;
  const float* brow = Wo + (size_t)(nbase + mr) * DM + 2 * hl;

  __builtin_prefetch(arow, 0, 3);
#pragma unroll
  for (int nt = 0; nt < 4; ++nt) __builtin_prefetch(brow + nt * 16 * DM, 0, 3);

  for (int k = 0; k < DM; k += 4) {
    v2f a = *(const v2f*)(arow + k);
#pragma unroll
    for (int nt = 0; nt < 4; ++nt) {
      v2f b = *(const v2f*)(brow + (size_t)nt * 16 * DM + k);
      acc[nt] = wmma4(a, b, acc[nt]);
    }
  }

#pragma unroll
  for (int nt = 0; nt < 4; ++nt) {
    const int col = nbase + nt * 16 + mr;
    const float bb = bo[col];
#pragma unroll
    for (int r = 0; r < 8; ++r) {
      const int row = mbase + r + 8 * hl;
      Y[(size_t)row * DM + col] = acc[nt][r] + bb + X[(size_t)row * DM + col];
    }
  }
}

// ---------------------------------------------------------------------------
extern "C" void kernel_launch(void* const* d_in, const int* in_sizes, int n_in,
                              void* d_out, int out_size, void* d_ws, size_t ws_size,
                              hipStream_t stream) {
  const float* X     = (const float*)d_in[0];
  const int*   sdiff = (const int*)  d_in[1];
  const int*   gmult = (const int*)  d_in[2];
  const float* Wq    = (const float*)d_in[3];
  const float* bq    = (const float*)d_in[4];
  const float* Wk    = (const float*)d_in[5];
  const float* bk    = (const float*)d_in[6];
  const float* Wv    = (const float*)d_in[7];
  const float* bv    = (const float*)d_in[8];
  const float* Wo    = (const float*)d_in[9];
  const float* bo    = (const float*)d_in[10];
  const float* rho   = (const float*)d_in[11];
  const float* ln_g  = (const float*)d_in[12];
  const float* ln_b  = (const float*)d_in[13];
  float* Y = (float*)d_out;

  const size_t MAT = (size_t)TOK * DM;   // 4096*512 floats
  float* Xn = (float*)d_ws;
  float* Q  = Xn + MAT;
  float* K  = Q + MAT;
  float* V  = K + MAT;
  float* O  = V + MAT;

  // 1) LayerNorm
  ln_kernel<<<TOK, 256, 0, stream>>>(X, ln_g, ln_b, Xn);

  // 2) QKV projections (z selects Wq/Wk/Wv)
  qkv_gemm<<<dim3(TOK / 128, DM / 64, 3), 256, 0, stream>>>(
      Xn, Wq, bq, Wk, bk, Wv, bv, Q, K, V);

  // 3) Attention: TDM-staged Q, WMMA logits, LDS softmax, LDS-staged PV
  const size_t lds_bytes = (size_t)SMEM_FLOATS * sizeof(float);
  attn_kernel<<<dim3(SEQ / 16, NH, NB), 256, lds_bytes, stream>>>(
      Q, K, V, sdiff, gmult, rho, O);

  // 4) Output projection + bias + residual
  out_gemm<<<dim3(TOK / 128, DM / 64), 256, 0, stream>>>(O, Wo, bo, X, Y);
}